// RecSysGNN_74363063763024
// MI455X (gfx1250) — compile-verified
//
#include <hip/hip_runtime.h>
#include <hip/hip_bf16.h>
#include <math.h>

typedef float v2f __attribute__((ext_vector_type(2)));
typedef float v8f __attribute__((ext_vector_type(8)));

constexpr int D = 64;           // feature dim (fixed by reference)
constexpr int BLOCK = 256;      // 8 wave32 per block
constexpr int WPB = 8;          // waves per block
constexpr int TILE = 16;        // rows (edges/nodes) per wave tile

#define WMMA_F32X4(a, b, c) \
  __builtin_amdgcn_wmma_f32_16x16x4_f32(false, (a), false, (b), (short)0, (c), false, false)

// Guaranteed hardware fp32 atomic add (non-returning, device scope, STOREcnt).
// Inline asm so we *know* the single-instruction L2 atomic is emitted rather
// than a CAS loop -- the scatter-add is this workload's bottleneck.
__device__ __forceinline__ void hw_atomic_add_f32(float* p, float v) {
  asm volatile("global_atomic_add_f32 %0, %1, off scope:SCOPE_DEV"
               :: "v"(p), "v"(v) : "memory");
}

// ---------------------------------------------------------------- utilities
__global__ __launch_bounds__(256) void k_zero(float* __restrict__ p, int n) {
  int i = blockIdx.x * blockDim.x + threadIdx.x;
  int s = gridDim.x * blockDim.x;
  for (; i < n; i += s) p[i] = 0.0f;
}

__global__ __launch_bounds__(256) void k_deg(const int* __restrict__ to,
                                             float* __restrict__ deg, int E) {
  int i = blockIdx.x * blockDim.x + threadIdx.x;
  int s = gridDim.x * blockDim.x;
  for (; i < E; i += s) hw_atomic_add_f32(&deg[to[i]], 1.0f);
}

__global__ __launch_bounds__(256) void k_dis(float* __restrict__ deg, int N) {
  int i = blockIdx.x * blockDim.x + threadIdx.x;
  int s = gridDim.x * blockDim.x;
  for (; i < N; i += s) {
    float d = deg[i];
    deg[i] = d > 0.0f ? 1.0f / sqrtf(d) : 0.0f;  // deg>0 -> max(deg,1)==deg
  }
}

__global__ __launch_bounds__(256) void k_norm(const int* __restrict__ frm,
                                              const int* __restrict__ to,
                                              const float* __restrict__ w,
                                              const float* __restrict__ dis,
                                              float* __restrict__ nrm, int E) {
  int i = blockIdx.x * blockDim.x + threadIdx.x;
  int s = gridDim.x * blockDim.x;
  for (; i < E; i += s) nrm[i] = dis[frm[i]] * dis[to[i]] * w[i];
}

__global__ __launch_bounds__(256) void k_copy(const float* __restrict__ src,
                                              float* __restrict__ dst, int n) {
  int i = blockIdx.x * blockDim.x + threadIdx.x;
  int s = gridDim.x * blockDim.x;
  for (; i < n; i += s) dst[i] = src[i];
}

// out = [emb0 (N*64)] ++ [concat(emb0, x) (N*128)]
__global__ __launch_bounds__(256) void k_out(const float* __restrict__ emb0,
                                             const float* __restrict__ x,
                                             float* __restrict__ out, int N) {
  int total = N * (D + 2 * D);
  int i = blockIdx.x * blockDim.x + threadIdx.x;
  int s = gridDim.x * blockDim.x;
  for (; i < total; i += s) {
    if (i < N * D) {
      out[i] = emb0[i];
    } else {
      int k = i - N * D;
      int n = k >> 7;           // /128
      int c = k & 127;
      out[i] = (c < D) ? emb0[n * D + c] : x[n * D + (c - D)];
    }
  }
}

// ------------------------------------------------- edge message + scatter
// msg = norm[e] * (xj@W1^T + (xj*xi)@W2^T + (b1+b2));  agg[to[e]] += msg
// One wave computes a 16-edge x 64-col tile with V_WMMA_F32_16X16X4_F32.
__global__ __launch_bounds__(256) void k_edge(const int* __restrict__ frm,
                                              const int* __restrict__ to,
                                              const float* __restrict__ nrm,
                                              const float* __restrict__ x,
                                              const float* __restrict__ W1l,
                                              const float* __restrict__ b1l,
                                              const float* __restrict__ W2l,
                                              const float* __restrict__ b2l,
                                              float* __restrict__ agg, int E) {
  __shared__ float sW1[D * D];   // 16 KB, row-major [out][in]
  __shared__ float sW2[D * D];   // 16 KB
  __shared__ float sB[D];        // fused bias b1+b2
  int tid = threadIdx.x;
  for (int i = tid; i < D * D; i += BLOCK) { sW1[i] = W1l[i]; sW2[i] = W2l[i]; }
  if (tid < D) sB[tid] = b1l[tid] + b2l[tid];
  __syncthreads();

  int wave = tid >> 5;
  int lane = tid & 31;
  int half = lane >> 4;          // 0: lanes 0-15, 1: lanes 16-31
  int lcol = lane & 15;
  int khoff = half * 2;          // A/B K-slot offset for upper half-wave

  int e0 = (blockIdx.x * WPB + wave) * TILE;
  if (e0 >= E) return;           // wave-uniform exit

  // A-matrix row for this lane is M = lane&15 (both halves), K varies by half
  int ea = e0 + lcol;
  if (ea >= E) ea = E - 1;       // clamp loads; stores guarded below
  const float* xj = x + (size_t)frm[ea] * D;
  const float* xi = x + (size_t)to[ea] * D;

  v8f acc[4] = {v8f{}, v8f{}, v8f{}, v8f{}};

  #pragma unroll
  for (int kk = 0; kk < 16; ++kk) {       // K in steps of 4
    int k = kk * 4 + khoff;
    v2f a1 = *(const v2f*)(xj + k);       // A fragment of xj
    v2f av = *(const v2f*)(xi + k);
    v2f a2 = a1 * av;                     // A fragment of xj*xi
    #pragma unroll
    for (int nt = 0; nt < 4; ++nt) {      // N in 4 tiles of 16
      // B[k][n] = W[n][k] (W row-major [out][in]); lane holds n = nt*16+lcol
      v2f b1v = *(const v2f*)(sW1 + (nt * 16 + lcol) * D + k);
      v2f b2v = *(const v2f*)(sW2 + (nt * 16 + lcol) * D + k);
      acc[nt] = WMMA_F32X4(a1, b1v, acc[nt]);
      acc[nt] = WMMA_F32X4(a2, b2v, acc[nt]);
    }
  }

  // D layout: VGPR r holds row m = r + 8*half, col n = nt*16 + (lane&15)
  #pragma unroll
  for (int r = 0; r < 8; ++r) {
    int m = r + half * 8;
    int e = e0 + m;
    if (e < E) {
      float nv = nrm[e];
      float* dst = agg + (size_t)to[e] * D;
      #pragma unroll
      for (int nt = 0; nt < 4; ++nt) {
        int col = nt * 16 + lcol;
        float v = nv * (acc[nt][r] + sB[col]);
        hw_atomic_add_f32(dst + col, v);
      }
    }
  }
}

// ------------------------------------------------- node update (per layer)
// xout = leaky_relu(agg + x@W1^T + b1, 0.01)
__global__ __launch_bounds__(256) void k_node(const float* __restrict__ x,
                                              const float* __restrict__ agg,
                                              const float* __restrict__ W1l,
                                              const float* __restrict__ b1l,
                                              float* __restrict__ xout, int N) {
  __shared__ float sW[D * D];
  __shared__ float sB[D];
  int tid = threadIdx.x;
  for (int i = tid; i < D * D; i += BLOCK) sW[i] = W1l[i];
  if (tid < D) sB[tid] = b1l[tid];
  __syncthreads();

  int wave = tid >> 5;
  int lane = tid & 31;
  int half = lane >> 4;
  int lcol = lane & 15;
  int khoff = half * 2;

  int n0 = (blockIdx.x * WPB + wave) * TILE;
  if (n0 >= N) return;

  int nr = n0 + lcol;
  if (nr >= N) nr = N - 1;
  const float* xr = x + (size_t)nr * D;

  v8f acc[4] = {v8f{}, v8f{}, v8f{}, v8f{}};

  #pragma unroll
  for (int kk = 0; kk < 16; ++kk) {
    int k = kk * 4 + khoff;
    v2f a = *(const v2f*)(xr + k);
    #pragma unroll
    for (int nt = 0; nt < 4; ++nt) {
      v2f b = *(const v2f*)(sW + (nt * 16 + lcol) * D + k);
      acc[nt] = WMMA_F32X4(a, b, acc[nt]);
    }
  }

  #pragma unroll
  for (int r = 0; r < 8; ++r) {
    int m = r + half * 8;
    int node = n0 + m;
    if (node < N) {
      const float* ag = agg + (size_t)node * D;
      float* xo = xout + (size_t)node * D;
      #pragma unroll
      for (int nt = 0; nt < 4; ++nt) {
        int col = nt * 16 + lcol;
        float v = acc[nt][r] + ag[col] + sB[col];
        xo[col] = v > 0.0f ? v : 0.01f * v;   // leaky_relu
      }
    }
  }
}

// --------------------------------------------------------------- launcher
extern "C" void kernel_launch(void* const* d_in, const int* in_sizes, int n_in,
                              void* d_out, int out_size, void* d_ws, size_t ws_size,
                              hipStream_t stream) {
  const int*   edge_index  = (const int*)d_in[0];   // (2,E) int32
  const float* edge_weight = (const float*)d_in[1]; // (E,)
  const float* emb         = (const float*)d_in[2]; // (N,64)
  const float* W1          = (const float*)d_in[3]; // (L,64,64)
  const float* b1          = (const float*)d_in[4]; // (L,64)
  const float* W2          = (const float*)d_in[5]; // (L,64,64)
  const float* b2          = (const float*)d_in[6]; // (L,64)

  const int E = in_sizes[1];
  const int N = in_sizes[2] / D;
  const int L = in_sizes[3] / (D * D);
  const int* frm = edge_index;
  const int* to  = edge_index + E;

  // workspace layout (floats): norm[E] deg[N] xA[N*D] xB[N*D] agg[N*D]
  float* ws   = (float*)d_ws;
  float* nrm  = ws;
  float* deg  = nrm + E;
  float* xA   = deg + N;
  float* xB   = xA + (size_t)N * D;
  float* agg  = xB + (size_t)N * D;

  auto cdiv = [](int a, int b) { return (a + b - 1) / b; };
  int gN   = min(cdiv(N, BLOCK), 2048);
  int gE   = min(cdiv(E, BLOCK), 4096);
  int gND  = min(cdiv(N * D, BLOCK), 4096);

  // symmetric-normalization coefficients (hoisted across layers)
  k_zero<<<gN, BLOCK, 0, stream>>>(deg, N);
  k_deg <<<gE, BLOCK, 0, stream>>>(to, deg, E);
  k_dis <<<gN, BLOCK, 0, stream>>>(deg, N);                 // deg -> dis in place
  k_norm<<<gE, BLOCK, 0, stream>>>(frm, to, edge_weight, deg, nrm, E);
  k_copy<<<gND, BLOCK, 0, stream>>>(emb, xA, N * D);

  float* xcur = xA;
  float* xnext = xB;
  const int blocksE = cdiv(cdiv(E, TILE), WPB);
  const int blocksN = cdiv(cdiv(N, TILE), WPB);
  for (int l = 0; l < L; ++l) {
    k_zero<<<gND, BLOCK, 0, stream>>>(agg, N * D);
    k_edge<<<blocksE, BLOCK, 0, stream>>>(frm, to, nrm, xcur,
                                          W1 + (size_t)l * D * D, b1 + (size_t)l * D,
                                          W2 + (size_t)l * D * D, b2 + (size_t)l * D,
                                          agg, E);
    k_node<<<blocksN, BLOCK, 0, stream>>>(xcur, agg,
                                          W1 + (size_t)l * D * D, b1 + (size_t)l * D,
                                          xnext, N);
    float* t = xcur; xcur = xnext; xnext = t;
  }

  int gOut = min(cdiv(N * 3 * D, BLOCK), 4096);
  k_out<<<gOut, BLOCK, 0, stream>>>(emb, xcur, (float*)d_out, N);
}